// MultiLoss_57234734187054
// MI455X (gfx1250) — compile-verified
//
#include <hip/hip_runtime.h>
#include <hip/hip_bf16.h>
#include <math.h>

#define B_DIM 512
#define D_DIM 16384
#define C_DIM 3
#define NT 256
#define NEG_RATIO 3

typedef __attribute__((ext_vector_type(4))) unsigned int u32x4;
typedef __attribute__((ext_vector_type(8))) int          i32x8;
typedef __attribute__((ext_vector_type(4))) int          i32x4;

// ---------------------------------------------------------------------------
// TDM: 1-D tile load, Global -> LDS.  n8 = number of 8-byte elements.
// D# layout per CDNA5 ISA ch.8:
//   group0: [1:0]=count=1, [63:32]=lds_addr, [120:64]=global_addr, [127:126]=type=2
//   group1: [17:16]=data_size(3 => 8B), [79:48]=tensor_dim0, [111:80]=tensor_dim1,
//           [127:112]=tile_dim0, [143:128]=tile_dim1, [207:160]=tensor_dim0_stride
// Toolchain here is clang-23 (therock-10.0 headers): 6-arg builtin form.
// ---------------------------------------------------------------------------
__device__ __forceinline__ void tdm_load_1d_to_lds(const void* gptr,
                                                   unsigned lds_byte_off,
                                                   unsigned n8) {
  unsigned long long ga = (unsigned long long)gptr;
  u32x4 g0;
  g0.x = 1u;                                        // count=1, user descriptor
  g0.y = lds_byte_off;                              // lds_addr (bytes)
  g0.z = (unsigned)(ga & 0xFFFFFFFFu);              // global_addr[31:0]
  g0.w = (unsigned)((ga >> 32) & 0x1FFFFFFu)        // global_addr[56:32]
       | (2u << 30);                                // type = 2 ("image")
  i32x8 g1;
  g1.s0 = (int)(3u << 16);                          // data_size = 8B; no mask/flags
  g1.s1 = (int)((n8 & 0xFFFFu) << 16);              // tensor_dim0[15:0]
  g1.s2 = (int)(((n8 >> 16) & 0xFFFFu) | (1u << 16)); // tensor_dim0[31:16] | tensor_dim1=1
  g1.s3 = (int)((n8 & 0xFFFFu) << 16);              // tensor_dim1[31:16]=0 | tile_dim0
  g1.s4 = 1;                                        // tile_dim1 = 1, tile_dim2 = 0
  g1.s5 = (int)n8;                                  // tensor_dim0_stride[31:0]
  g1.s6 = 0;
  g1.s7 = 0;
  i32x4 z4 = (i32x4)0;
  i32x8 z8 = (i32x8)0;
  __builtin_amdgcn_tensor_load_to_lds(g0, g1, z4, z4, z8, 0);
}

// ---------------------------------------------------------------------------
// K1: per-row matching. blockIdx.x = batch row b. 256 threads (8 waves).
// Dynamic LDS: [0,131072) default_bar copy; then 256 floats + 256 ints scratch.
// ---------------------------------------------------------------------------
__global__ __launch_bounds__(NT)
void k_multiloss_match(const float* __restrict__ loc_pred,
                       const float* __restrict__ conf_pred,
                       const float* __restrict__ targets,
                       const float* __restrict__ default_bar,
                       float* __restrict__ lc_out,
                       float* __restrict__ meta_loc,
                       float* __restrict__ meta_cep,
                       int*   __restrict__ meta_np) {
  extern __shared__ char smem[];
  float* s_def = (float*)smem;                              // 2*D_DIM floats
  float* redf  = (float*)(smem + 2u * D_DIM * sizeof(float));
  int*   redi  = (int*)(smem + 2u * D_DIM * sizeof(float) + NT * sizeof(float));

  const int b   = blockIdx.x;
  const int tid = threadIdx.x;
  const float t0 = targets[b * 3 + 0];
  const float t1 = targets[b * 3 + 1];
  const int  lab = (int)targets[b * 3 + 2];

  // Stage default_bar (128KB) into LDS: each wave DMAs its 16KB slice via TDM.
  {
    unsigned wave = (unsigned)tid >> 5;                     // wave32
    tdm_load_1d_to_lds(default_bar + (size_t)wave * 4096u,  // 4096 floats = 16KB
                       wave * 16384u, 2048u);               // 2048 x 8B
    __builtin_amdgcn_s_wait_tensorcnt(0);
  }
  __syncthreads();

  // ---- Phase A: argmax of IoU overlap (first-index tie-break) ----
  float bestv = -1.0f;
  int   besti = 0;
  for (int d = tid; d < D_DIM; d += NT) {
    float c = s_def[2 * d], l = s_def[2 * d + 1];
    float ds = c - l * 0.5f, de = c + l * 0.5f;
    float inter = fmaxf(fminf(t1, de) - fmaxf(t0, ds), 0.0f);
    float uni   = (t1 - t0) + (de - ds) - inter;
    float ov    = inter / uni;
    if (ov > bestv) { bestv = ov; besti = d; }              // keeps smallest d
  }
  redf[tid] = bestv; redi[tid] = besti;
  __syncthreads();
  for (int s = NT / 2; s > 0; s >>= 1) {
    if (tid < s) {
      float v2 = redf[tid + s]; int i2 = redi[tid + s];
      if (v2 > redf[tid] || (v2 == redf[tid] && i2 < redi[tid])) {
        redf[tid] = v2; redi[tid] = i2;
      }
    }
    __syncthreads();
  }
  const int best_prior = redi[0];
  __syncthreads();

  // ---- Phase B: conf_t, smooth-L1 loc loss, ce, masked lc ----
  const float m_c = (t0 + t1) * 0.5f;
  const float m_l = t1 - t0;
  float loc_acc = 0.0f, cep_acc = 0.0f;
  int   np_acc  = 0;
  for (int d = tid; d < D_DIM; d += NT) {
    float c = s_def[2 * d], l = s_def[2 * d + 1];
    float ds = c - l * 0.5f, de = c + l * 0.5f;
    float inter = fmaxf(fminf(t1, de) - fmaxf(t0, ds), 0.0f);
    float uni   = (t1 - t0) + (de - ds) - inter;
    float ov    = inter / uni;
    bool hit    = (ov >= 0.5f) || (d == best_prior);        // bto >= NEG_THRESHOLD
    int conf_t  = (lab == 0) ? 1 : (hit ? (lab + 1) : 0);
    bool pos    = conf_t > 0;

    if (conf_t > 1) {                                       // loc_pos
      size_t li = ((size_t)b * D_DIM + (size_t)d) * 2u;
      float p0 = loc_pred[li], p1 = loc_pred[li + 1];
      float lt0 = (m_c - c) / l;
      float lt1 = logf(m_l / l);
      float d0 = p0 - lt0, d1 = p1 - lt1;
      float a0 = fabsf(d0), a1 = fabsf(d1);
      loc_acc += (a0 < 1.0f) ? 0.5f * d0 * d0 : a0 - 0.5f;
      loc_acc += (a1 < 1.0f) ? 0.5f * d1 * d1 : a1 - 0.5f;
    }

    size_t ci = ((size_t)b * D_DIM + (size_t)d) * 3u;
    float x0 = conf_pred[ci], x1 = conf_pred[ci + 1], x2 = conf_pred[ci + 2];
    float m  = fmaxf(fmaxf(x0, x1), x2);
    float se = expf(x0 - m) + expf(x1 - m) + expf(x2 - m);
    float lse = m + logf(se);
    int tc = conf_t - 1; if (tc < 0) tc = 0;
    float g  = (tc == 0) ? x0 : ((tc == 1) ? x1 : x2);
    float ce = lse - g;
    if (pos) { cep_acc += ce; np_acc++; }
    lc_out[(size_t)b * D_DIM + (size_t)d] = pos ? 0.0f : ce;
  }

  // Deterministic block tree reductions (no float atomics).
  redf[tid] = loc_acc; __syncthreads();
  for (int s = NT / 2; s > 0; s >>= 1) { if (tid < s) redf[tid] += redf[tid + s]; __syncthreads(); }
  float loc_sum = redf[0]; __syncthreads();
  redf[tid] = cep_acc; __syncthreads();
  for (int s = NT / 2; s > 0; s >>= 1) { if (tid < s) redf[tid] += redf[tid + s]; __syncthreads(); }
  float cep_sum = redf[0]; __syncthreads();
  redi[tid] = np_acc; __syncthreads();
  for (int s = NT / 2; s > 0; s >>= 1) { if (tid < s) redi[tid] += redi[tid + s]; __syncthreads(); }
  if (tid == 0) {
    meta_loc[b] = loc_sum;
    meta_cep[b] = cep_sum;
    meta_np[b]  = redi[0];
  }
}

// ---------------------------------------------------------------------------
// K2: per-row hard-negative mining via 4-pass byte radix-select (lc >= 0, so
// IEEE bit pattern order == value order). Then:
//   conf_row = ce_pos + sum_{lc > v} lc + (k - count_{lc > v}) * v
// which is exact incl. stable-sort tie semantics (ties add v each; v==0 => 0).
// ---------------------------------------------------------------------------
__global__ __launch_bounds__(NT)
void k_multiloss_select(const float* __restrict__ lc,
                        const float* __restrict__ meta_cep,
                        const int*   __restrict__ meta_np,
                        float* __restrict__ meta_conf) {
  __shared__ unsigned hist[256];
  __shared__ float    redf[NT];
  __shared__ unsigned redu[NT];
  __shared__ unsigned s_prefix;
  __shared__ int      s_need;

  const int b   = blockIdx.x;
  const int tid = threadIdx.x;
  const float* row = lc + (size_t)b * D_DIM;

  int np = meta_np[b];
  long long kk = (long long)NEG_RATIO * (long long)np;
  if (kk > D_DIM - 1) kk = D_DIM - 1;
  const int k = (int)kk;
  const float cep = meta_cep[b];
  if (k <= 0) { if (tid == 0) meta_conf[b] = cep; return; }

  if (tid == 0) { s_prefix = 0u; s_need = k; }
  __syncthreads();

  for (int pass = 3; pass >= 0; --pass) {
    hist[tid] = 0u;
    __syncthreads();
    const unsigned prefix = s_prefix;
    const int shift = pass * 8;
    const unsigned himask = (pass == 3) ? 0u : (0xFFFFFFFFu << (shift + 8));
    for (int d = tid; d < D_DIM; d += NT) {
      unsigned u = __float_as_uint(row[d]);
      if ((u & himask) == prefix) atomicAdd(&hist[(u >> shift) & 255u], 1u);
    }
    __syncthreads();
    if (tid == 0) {
      int need = s_need;
      unsigned cum = 0; int chosen = 0;
      for (int bin = 255; bin >= 0; --bin) {
        unsigned h = hist[bin];
        if ((unsigned)need <= cum + h) { chosen = bin; need -= (int)cum; break; }
        cum += h;
      }
      s_prefix = prefix | ((unsigned)chosen << shift);
      s_need   = need;
    }
    __syncthreads();
  }

  const unsigned vbits = s_prefix;
  unsigned cnt = 0; float sum = 0.0f;
  for (int d = tid; d < D_DIM; d += NT) {
    float val = row[d];
    unsigned u = __float_as_uint(val);
    if (u > vbits) { cnt++; sum += val; }
  }
  redu[tid] = cnt; redf[tid] = sum;
  __syncthreads();
  for (int s = NT / 2; s > 0; s >>= 1) {
    if (tid < s) { redu[tid] += redu[tid + s]; redf[tid] += redf[tid + s]; }
    __syncthreads();
  }
  if (tid == 0) {
    float v = __uint_as_float(vbits);
    meta_conf[b] = cep + redf[0] + (float)(k - (int)redu[0]) * v;
  }
}

// ---------------------------------------------------------------------------
// K3: deterministic final reduction over 512 rows, divide by N.
// ---------------------------------------------------------------------------
__global__ __launch_bounds__(NT)
void k_multiloss_final(const float* __restrict__ meta_loc,
                       const float* __restrict__ meta_conf,
                       const int*   __restrict__ meta_np,
                       float* __restrict__ out) {
  __shared__ float rl[NT], rc[NT];
  __shared__ int   rn[NT];
  const int tid = threadIdx.x;
  float l = 0.0f, c = 0.0f; int n = 0;
  for (int b = tid; b < B_DIM; b += NT) { l += meta_loc[b]; c += meta_conf[b]; n += meta_np[b]; }
  rl[tid] = l; rc[tid] = c; rn[tid] = n;
  __syncthreads();
  for (int s = NT / 2; s > 0; s >>= 1) {
    if (tid < s) { rl[tid] += rl[tid + s]; rc[tid] += rc[tid + s]; rn[tid] += rn[tid + s]; }
    __syncthreads();
  }
  if (tid == 0) {
    float N = (float)rn[0];
    out[0] = rl[0] / N;
    out[1] = rc[0] / N;
  }
}

// ---------------------------------------------------------------------------
extern "C" void kernel_launch(void* const* d_in, const int* in_sizes, int n_in,
                              void* d_out, int out_size, void* d_ws, size_t ws_size,
                              hipStream_t stream) {
  const float* loc_pred    = (const float*)d_in[0];
  const float* conf_pred   = (const float*)d_in[1];
  const float* targets     = (const float*)d_in[2];
  const float* default_bar = (const float*)d_in[3];
  float* out = (float*)d_out;

  char* ws = (char*)d_ws;
  float* lc = (float*)ws;
  size_t off = (size_t)B_DIM * D_DIM * sizeof(float);      // 32 MB for lc
  float* meta_loc  = (float*)(ws + off);
  float* meta_cep  = (float*)(ws + off + 1u * B_DIM * sizeof(float));
  float* meta_conf = (float*)(ws + off + 2u * B_DIM * sizeof(float));
  int*   meta_np   = (int*)  (ws + off + 3u * B_DIM * sizeof(float));

  const size_t smem1 = 2u * D_DIM * sizeof(float)          // staged default_bar
                     + NT * sizeof(float) + NT * sizeof(int);
  (void)hipFuncSetAttribute((const void*)k_multiloss_match,
                            hipFuncAttributeMaxDynamicSharedMemorySize,
                            (int)smem1);

  k_multiloss_match<<<B_DIM, NT, smem1, stream>>>(loc_pred, conf_pred, targets,
                                                  default_bar, lc, meta_loc,
                                                  meta_cep, meta_np);
  k_multiloss_select<<<B_DIM, NT, 0, stream>>>(lc, meta_cep, meta_np, meta_conf);
  k_multiloss_final<<<1, NT, 0, stream>>>(meta_loc, meta_conf, meta_np, out);
}